// GQM_16398185136153
// MI455X (gfx1250) — compile-verified
//
#include <hip/hip_runtime.h>

// CDNA5 / gfx1250: wave32, fp32 WMMA (V_WMMA_F32_16X16X4_F32) + Tensor Data Mover.
typedef __attribute__((ext_vector_type(2))) float v2f;
typedef __attribute__((ext_vector_type(8))) float v8f;
typedef __attribute__((ext_vector_type(4))) unsigned int u32x4;
typedef __attribute__((ext_vector_type(8))) int i32x8;
typedef __attribute__((ext_vector_type(4))) int i32x4;

#define BS_  8
#define N_   64
#define H_   512
#define L_   5
#define NPIX 196

__device__ __forceinline__ float lrelu(float x) { return x > 0.f ? x : 0.01f * x; }

// ---------------------------------------------------------------------------
// TDM 2-D tile load: global (tile_d1 rows x tile_d0 elems, row stride stride0,
// all in 4-byte units) -> contiguous LDS at byte offset lds_off.
// Descriptor layout per CDNA5 ISA (groups 2/3 zero => 2D).
// This toolchain exposes the 6-arg builtin: (g0, g1, g2, g3, g4, cpol).
__device__ __forceinline__ void tdm_load_2d(unsigned lds_off, const float* gaddr,
                                            unsigned tile_d0, unsigned tile_d1,
                                            unsigned tensor_d0, unsigned tensor_d1,
                                            unsigned stride0) {
    unsigned long long ga = (unsigned long long)(__UINTPTR_TYPE__)gaddr;
    u32x4 g0;
    g0[0] = 1u;                                             // count=1, user mode
    g0[1] = lds_off;                                        // lds_addr (bytes)
    g0[2] = (unsigned)(ga & 0xffffffffu);                   // global_addr[31:0]
    g0[3] = (unsigned)((ga >> 32) & 0x01ffffffu) | (2u << 30); // addr[56:32] | type=2
    i32x8 g1;
    g1[0] = (int)(2u << 16);                                // data_size = 4 bytes
    g1[1] = (int)((tensor_d0 & 0xffffu) << 16);             // tensor_dim0[15:0]
    g1[2] = (int)(((tensor_d0 >> 16) & 0xffffu) | ((tensor_d1 & 0xffffu) << 16));
    g1[3] = (int)(((tensor_d1 >> 16) & 0xffffu) | ((tile_d0 & 0xffffu) << 16));
    g1[4] = (int)(tile_d1 & 0xffffu);                       // tile_dim1 (tile_dim2=0)
    g1[5] = (int)stride0;                                   // tensor_dim0_stride[31:0]
    g1[6] = 0;
    g1[7] = 0;                                              // dim1_stride unused (2D)
    i32x4 z4 = (i32x4){0, 0, 0, 0};
    i32x8 z8 = (i32x8){0, 0, 0, 0, 0, 0, 0, 0};
    __builtin_amdgcn_tensor_load_to_lds(g0, g1, z4, z4, z8, 0);
}

// ---------------------------------------------------------------------------
// edge_init[b,h] = mean over 14x14 of img_featmap[b,h,:,:]
__global__ void k_edge_init(const float* __restrict__ img, float* __restrict__ edge_init) {
    int t = blockIdx.x * blockDim.x + threadIdx.x;
    if (t >= BS_ * H_) return;
    const float* p = img + (size_t)t * NPIX;
    float s = 0.f;
    for (int q = 0; q < NPIX; ++q) s += p[q];
    edge_init[t] = s * (1.0f / NPIX);
}

// denom[b,i] = 1 + sum_j mask[b,i,j]
__global__ void k_denom(const int* __restrict__ mask, float* __restrict__ denom) {
    int t = blockIdx.x * blockDim.x + threadIdx.x;
    if (t >= BS_ * N_) return;
    const int* mp = mask + (size_t)t * N_;
    int s = 0;
    for (int j = 0; j < N_; ++j) s += mp[j];
    denom[t] = (float)s + 1.0f;
}

// Layer-0 algebraic collapse: em0[b,o] = edge_init[b,:] . edge_W[0][o,:]  (no bias)
__global__ void k_em0(const float* __restrict__ edge_init, const float* __restrict__ edge_W,
                      float* __restrict__ em0) {
    int t = blockIdx.x * blockDim.x + threadIdx.x;
    if (t >= BS_ * H_) return;
    int b = t / H_, o = t % H_;
    const float* e = edge_init + (size_t)b * H_;
    const float* w = edge_W + (size_t)o * H_;
    float s = 0.f;
    for (int h = 0; h < H_; ++h) s += e[h] * w[h];
    em0[t] = s;
}

// ---------------------------------------------------------------------------
// nm[r,o] = node_state[r,:] . node_W[t][o,:] + node_b[t][o]   (512x512, K=512)
__global__ void k_nm(const float* __restrict__ node_state, const float* __restrict__ node_W,
                     const float* __restrict__ node_b, float* __restrict__ nm, int layer) {
    const int rt = blockIdx.x >> 2;
    const int og = blockIdx.x & 3;
    const int tid = threadIdx.x;
    const int w = tid >> 5;
    const int l = tid & 31;
    const int lm = l & 15, lh = l >> 4;
    const int row0 = rt * 16;
    const int o0 = og * 128 + w * 16;
    const float* Wt = node_W + (size_t)layer * H_ * H_;

    v8f acc = (v8f){0.f,0.f,0.f,0.f,0.f,0.f,0.f,0.f};
    const int arow = row0 + lm;
    const int bo   = o0 + lm;
    for (int k = 0; k < H_; k += 4) {
        const int kk = k + (lh << 1);
        v2f a  = *reinterpret_cast<const v2f*>(&node_state[(size_t)arow * H_ + kk]);
        v2f bf = *reinterpret_cast<const v2f*>(&Wt[(size_t)bo * H_ + kk]);
        acc = __builtin_amdgcn_wmma_f32_16x16x4_f32(false, a, false, bf, (short)0, acc,
                                                    false, false);
    }
    const float bias = node_b[layer * H_ + o0 + lm];
#pragma unroll
    for (int r = 0; r < 8; ++r) {
        const int m = r + (lh << 3);
        nm[(size_t)(row0 + m) * H_ + o0 + lm] = acc[r] + bias;
    }
}

// ---------------------------------------------------------------------------
// Fused per-layer edge kernel. Block = (b, i-pair): owns 128 edge rows (2 i x 64 j).
// TDM stages the old edge rows (128x512 f32 = 256 KB) into LDS once (in-place update
// is then race-free) and double-buffers 64x32 W tiles with a 1-deep TDM pipeline.
// Epilogue: leaky_relu(nm_i+nm_j+em+eb) -> edge_state, in-block agg over ALL j
// (shfl + LDS partials, no atomics), fused node_state update.
__global__ void k_edge(const float* __restrict__ nm, const float* __restrict__ edge_W,
                       const float* __restrict__ edge_b, const float* __restrict__ em0,
                       const float* __restrict__ denom, const int* __restrict__ mask,
                       float* __restrict__ edge_state, float* __restrict__ node_state,
                       int layer) {
    extern __shared__ float smem[];
    float* As    = smem;                        // 128 x 512 (full-K A tile), LDS off 0
    float* Bs0   = smem + 128 * H_;             // 64 x 32 buffer 0
    float* Bs1   = Bs0 + 64 * 32;               // 64 x 32 buffer 1
    float* agg_s = Bs1 + 64 * 32;               // 8 x 64
    const unsigned AS_OFF  = 0u;
    const unsigned BS_OFF0 = 128u * H_ * 4u;
    const unsigned BS_OFF1 = BS_OFF0 + 64u * 32u * 4u;

    const int blk = blockIdx.x;
    const int b = blk >> 5;
    const int p = blk & 31;
    const int tid = threadIdx.x;
    const int w = tid >> 5;
    const int l = tid & 31;
    const int lh = l >> 4;
    const int lm = l & 15;

    const size_t ebase = (size_t)(b * N_ + 2 * p) * N_;   // 128 contiguous edge rows
    const float* Wt = edge_W + (size_t)layer * H_ * H_;
    const float* eb = edge_b + layer * H_;

    const int erow  = 16 * w + lm;
    const int il    = w >> 2;
    const int i     = 2 * p + il;
    const int jbase = 16 * (w & 3);

    // --- TDM prologue: A tile + first W chunk (chunk c = oc*16 + kc) ---
    if (layer != 0 && tid < 32) {
        tdm_load_2d(AS_OFF, edge_state + ebase * H_, 512, 128, 512, 128, 512);
        tdm_load_2d(BS_OFF0, Wt, 32, 64, 32, 64, 512);     // oc=0, kc=0
    }

    for (int oc = 0; oc < 8; ++oc) {
        const int o0 = oc * 64;
        v8f acc[4];
#pragma unroll
        for (int t = 0; t < 4; ++t) acc[t] = (v8f){0.f,0.f,0.f,0.f,0.f,0.f,0.f,0.f};

        if (layer != 0) {
            for (int kc = 0; kc < 16; ++kc) {
                const int c = oc * 16 + kc;                // global chunk id
                if (tid < 32) {
                    if (c + 1 < 128) {                     // prefetch next W chunk
                        const int noc = (c + 1) >> 4, nkc = (c + 1) & 15;
                        tdm_load_2d(((c + 1) & 1) ? BS_OFF1 : BS_OFF0,
                                    Wt + (size_t)(noc * 64) * H_ + nkc * 32,
                                    32, 64, 32, 64, 512);
                        __builtin_amdgcn_s_wait_tensorcnt(1);  // current chunk resident
                    } else {
                        __builtin_amdgcn_s_wait_tensorcnt(0);
                    }
                }
                __syncthreads();
                const float* Bs = (c & 1) ? Bs1 : Bs0;
#pragma unroll
                for (int ks = 0; ks < 8; ++ks) {
                    const int ak = kc * 32 + ks * 4 + (lh << 1);
                    v2f a = *reinterpret_cast<const v2f*>(&As[erow * H_ + ak]);
#pragma unroll
                    for (int ot = 0; ot < 4; ++ot) {
                        const int bo = 16 * ot + lm;
                        v2f bf = *reinterpret_cast<const v2f*>(&Bs[bo * 32 + ks * 4 + (lh << 1)]);
                        acc[ot] = __builtin_amdgcn_wmma_f32_16x16x4_f32(
                            false, a, false, bf, (short)0, acc[ot], false, false);
                    }
                }
                __syncthreads();   // all waves done with this buffer before next TDM lands
            }
        }

        // ---- epilogue: activation, edge store, in-block agg over j ----
        float aggv[4];
#pragma unroll
        for (int ot = 0; ot < 4; ++ot) {
            const int o = o0 + 16 * ot + lm;
            const float base = (layer == 0) ? em0[b * H_ + o] : 0.f;
            const float nmi  = nm[(size_t)(b * N_ + i) * H_ + o];
            const float ebo  = eb[o];
            float s = 0.f;
#pragma unroll
            for (int r = 0; r < 8; ++r) {
                const int m = r + (lh << 3);
                const int j = jbase + m;
                const float nmj = nm[(size_t)(b * N_ + j) * H_ + o];
                const float mv  = (float)mask[(b * N_ + i) * N_ + j];
                float v = acc[ot][r] + base + nmi + nmj + ebo;
                v = lrelu(v);
                edge_state[((size_t)(b * N_ + i) * N_ + j) * H_ + o] = v;
                s += v * nmj * mv;
            }
            s += __shfl_xor(s, 16);
            aggv[ot] = s;
        }
        if (lh == 0) {
#pragma unroll
            for (int ot = 0; ot < 4; ++ot) agg_s[w * 64 + 16 * ot + lm] = aggv[ot];
        }
        __syncthreads();
        if (tid < 128) {
            const int il2 = tid >> 6;
            const int ol  = tid & 63;
            const float s = agg_s[(il2 * 4 + 0) * 64 + ol] + agg_s[(il2 * 4 + 1) * 64 + ol] +
                            agg_s[(il2 * 4 + 2) * 64 + ol] + agg_s[(il2 * 4 + 3) * 64 + ol];
            const int ii = 2 * p + il2;
            const int o  = o0 + ol;
            const float nmv = nm[(size_t)(b * N_ + ii) * H_ + o];
            const float nf  = (nmv + s) / denom[b * N_ + ii];
            node_state[(size_t)(b * N_ + ii) * H_ + o] += lrelu(nf);
        }
        __syncthreads();
    }
}

// ---------------------------------------------------------------------------
// out[b,i,j,0:512] = node[b,i,:]; out[b,i,j,512:1024] = node[b,j,:]. Pure BW kernel.
__global__ void k_out(const float* __restrict__ node_state, float* __restrict__ out) {
    size_t q = (size_t)blockIdx.x * blockDim.x + threadIdx.x;
    int c4 = (int)(q & 255);
    size_t r = q >> 8;
    int j = (int)(r & 63); r >>= 6;
    int i = (int)(r & 63);
    int b = (int)(r >> 6);
    int c = c4 << 2;
    float4 v;
    if (c < H_) v = *reinterpret_cast<const float4*>(&node_state[(size_t)(b * N_ + i) * H_ + c]);
    else        v = *reinterpret_cast<const float4*>(&node_state[(size_t)(b * N_ + j) * H_ + c - H_]);
    reinterpret_cast<float4*>(out)[q] = v;
}

// ---------------------------------------------------------------------------
extern "C" void kernel_launch(void* const* d_in, const int* in_sizes, int n_in,
                              void* d_out, int out_size, void* d_ws, size_t ws_size,
                              hipStream_t stream) {
    const float* roi    = (const float*)d_in[0];
    const float* img    = (const float*)d_in[1];
    const float* node_W = (const float*)d_in[2];
    const float* node_b = (const float*)d_in[3];
    const float* edge_W = (const float*)d_in[4];
    const float* edge_b = (const float*)d_in[5];
    const int*   mask   = (const int*)d_in[6];
    float* out = (float*)d_out;

    char* ws = (char*)d_ws;
    float* node_state = (float*)(ws + 0);                         // 1 MiB
    float* nm         = (float*)(ws + (1u << 20));                // 1 MiB
    float* edge_init  = (float*)(ws + (2u << 20));                // 16 KB
    float* em0        = (float*)(ws + (2u << 20) + (1u << 16));   // 16 KB
    float* denomb     = (float*)(ws + (2u << 20) + (2u << 16));   // 2 KB
    float* edge_state = (float*)(ws + (3u << 20));                // 64 MiB

    (void)hipMemcpyAsync(node_state, roi, (size_t)BS_ * N_ * H_ * sizeof(float),
                         hipMemcpyDeviceToDevice, stream);
    k_edge_init<<<(BS_ * H_ + 255) / 256, 256, 0, stream>>>(img, edge_init);
    k_denom<<<(BS_ * N_ + 255) / 256, 256, 0, stream>>>(mask, denomb);
    k_em0<<<(BS_ * H_ + 255) / 256, 256, 0, stream>>>(edge_init, edge_W, em0);

    const size_t lds_bytes = (128 * H_ + 2 * 64 * 32 + 8 * 64) * sizeof(float); // ~274 KB
    for (int t = 0; t < L_; ++t) {
        k_nm<<<128, 256, 0, stream>>>(node_state, node_W, node_b, nm, t);
        k_edge<<<BS_ * 32, 256, lds_bytes, stream>>>(nm, edge_W, edge_b, em0, denomb, mask,
                                                     edge_state, node_state, t);
    }
    k_out<<<(BS_ * N_ * N_ * (2 * H_ / 4)) / 256, 256, 0, stream>>>(node_state, out);
}